// TT_81303730913377
// MI455X (gfx1250) — compile-verified
//
#include <hip/hip_runtime.h>

typedef float v2f __attribute__((ext_vector_type(2)));
typedef float v8f __attribute__((ext_vector_type(8)));
typedef int   v4i __attribute__((ext_vector_type(4)));

#define NBATCH 4096
#define HID    64
#define CH     3
#define OUT_D  10
#define SITES  196
#define MIDS   194            // SITES-2 middle sites
#define WPITCH 68             // padded K pitch (bank-conflict-free, 8B aligned)
#define WROWS  128            // j = i*64 + l
#define WBLK   (WROWS * WPITCH)        // 8704 floats per (c, site)
#define WP_TOTAL (3LL * MIDS * WBLK)   // repacked weights in ws
#define XT_OFF   WP_TOTAL
#define XT_TOTAL (1LL * SITES * 2 * NBATCH)

#define BLOCK_THREADS 64               // 2 waves; each wave owns 32 batch rows
#define ROWS_PER_WAVE 32
#define ROWS_PER_WG   64
#define ASYNC_OPS_PER_THREAD ((WBLK / 4) / BLOCK_THREADS)   // 2176/64 = 34

#if defined(__has_builtin)
#if __has_builtin(__builtin_amdgcn_global_load_async_to_lds_b128) && \
    __has_builtin(__builtin_amdgcn_s_wait_asynccnt)
#define ASYNC_OK 1
#endif
#endif
#ifndef ASYNC_OK
#define ASYNC_OK 0
#endif

// ---------------------------------------------------------------------------
// One-time repack: w_mid (s,r,i,l,c) -> wp[c][s][j=(i,l)][r] (pitch 68, pads=0)
//                  x (n,i,s)         -> xT[s][i][n]
// ---------------------------------------------------------------------------
__global__ void TT_repack_kernel(const float* __restrict__ x,
                                 const float* __restrict__ w_mid,
                                 float* __restrict__ ws) {
  long long idx = (long long)blockIdx.x * blockDim.x + threadIdx.x;
  const long long total = WP_TOTAL + XT_TOTAL;
  if (idx >= total) return;
  if (idx < WP_TOTAL) {
    int t  = (int)idx;
    int c  = t / (MIDS * WBLK);
    int rm = t % (MIDS * WBLK);
    int s  = rm / WBLK;
    int rr = rm % WBLK;
    int j  = rr / WPITCH;
    int r  = rr % WPITCH;
    float v = 0.0f;
    if (r < HID) {
      int i = j >> 6;       // i in {0,1}
      int l = j & 63;       // hidden output index
      v = w_mid[((((long long)s * HID + r) * 2 + i) * HID + l) * CH + c];
    }
    ws[t] = v;
  } else {
    int t = (int)(idx - WP_TOTAL);
    int s = t / (2 * NBATCH);
    int i = (t / NBATCH) & 1;
    int n = t % NBATCH;
    ws[XT_OFF + t] = x[(long long)n * (2 * SITES) + i * SITES + s];
  }
}

// ---------------------------------------------------------------------------
// Main chain kernel: block = 2 waves, one channel, 64 batch rows (32/wave).
// Per site per wave: P = Y(32x64) * Wcat(64x128) via V_WMMA_F32_16X16X4_F32
// (256 WMMAs, each LDS B-fragment reused for 2 M-tiles), then
// Y_new = x0 .* P[:, :64] + x1 .* P[:, 64:].
// W double-buffered; site s+1 prefetched with async global->LDS copies.
// ---------------------------------------------------------------------------
__launch_bounds__(BLOCK_THREADS)
__global__ void TT_mps_chain_kernel(const float* __restrict__ ws,
                                    const float* __restrict__ w_first,
                                    const float* __restrict__ w_last,
                                    float* __restrict__ out) {
  __shared__ float Wbuf[2][WBLK];                        // 69632 B
  __shared__ float Ybuf[2][ROWS_PER_WAVE * WPITCH];      // 17408 B (per-wave tile)
  __shared__ float xbuf[2][2 * ROWS_PER_WG];             // 1024 B

  const int tid    = threadIdx.x;
  const int wave   = tid >> 5;
  const int lane   = tid & 31;
  const int lane16 = lane & 15;
  const int half   = lane >> 4;           // 0: lanes 0-15, 1: lanes 16-31
  const int koff   = half * 2;            // A/B fragment K offset per lane half

  const int c    = blockIdx.x % 3;
  const int tile = blockIdx.x / 3;
  const int n0   = tile * ROWS_PER_WG;

  const float* __restrict__ xT = ws + XT_OFF;

  // ---- site 0: Y[m][r] = x0*wf[0,r,c] + x1*wf[1,r,c]
  for (int idx = tid; idx < ROWS_PER_WG * HID; idx += BLOCK_THREADS) {
    int mg = idx >> 6;                    // 0..63 row in WG tile
    int r  = idx & 63;
    int n  = n0 + mg;
    float x0 = xT[0 * NBATCH + n];        // s=0, i=0
    float x1 = xT[1 * NBATCH + n];        // s=0, i=1
    float v  = x0 * w_first[(0 * HID + r) * CH + c]
             + x1 * w_first[(1 * HID + r) * CH + c];
    Ybuf[mg >> 5][(mg & 31) * WPITCH + r] = v;
  }

#if ASYNC_OK
  // prologue: kick off async staging of site 1 into Wbuf[0]
  {
    const float4* __restrict__ src = (const float4*)(ws + (long long)(c * MIDS) * WBLK);
    float4* dst = (float4*)Wbuf[0];
    #pragma unroll 2
    for (int t = tid; t < WBLK / 4; t += BLOCK_THREADS) {
      __builtin_amdgcn_global_load_async_to_lds_b128(
          (v4i*)(src + t),
          (__attribute__((address_space(3))) v4i*)(dst + t), 0, 0);
    }
  }
#endif

  // ---- middle sites 1..194
  for (int s = 1; s <= MIDS; ++s) {
    const int cur = (s - 1) & 1;
    const int sb  = s & 1;
    __syncthreads();   // all compute of site s-1 done (Wbuf[sb], xbuf[sb] free)

#if ASYNC_OK
    if (s < MIDS) {    // prefetch site s+1 into the other buffer
      const float4* __restrict__ src =
          (const float4*)(ws + (long long)(c * MIDS + s) * WBLK);
      float4* dst = (float4*)Wbuf[sb];
      #pragma unroll 2
      for (int t = tid; t < WBLK / 4; t += BLOCK_THREADS) {
        __builtin_amdgcn_global_load_async_to_lds_b128(
            (v4i*)(src + t),
            (__attribute__((address_space(3))) v4i*)(dst + t), 0, 0);
      }
    }
#else
    {                  // fallback: synchronous staging of site s into Wbuf[cur]
      const float4* __restrict__ src =
          (const float4*)(ws + (long long)(c * MIDS + (s - 1)) * WBLK);
      float4* dst = (float4*)Wbuf[cur];
      #pragma unroll 2
      for (int t = tid; t < WBLK / 4; t += BLOCK_THREADS) dst[t] = src[t];
    }
#endif
    // stage x pair for site s (double-buffered)
    for (int t = tid; t < 2 * ROWS_PER_WG; t += BLOCK_THREADS)
      xbuf[sb][t] = xT[((long long)s * 2 + (t >> 6)) * NBATCH + n0 + (t & 63)];

#if ASYNC_OK
    // site-s batch was issued one iteration ago; in-order completion means
    // waiting down to the just-issued batch guarantees site s has landed.
    if (s < MIDS) __builtin_amdgcn_s_wait_asynccnt(ASYNC_OPS_PER_THREAD);
    else          __builtin_amdgcn_s_wait_asynccnt(0);
#endif
    __syncthreads();   // all waves' staging of site s visible

    v8f acc[2][8] = {};
    const float* __restrict__ Yw = Ybuf[wave];
    const float* __restrict__ Wc = Wbuf[cur];

    for (int kk = 0; kk < 16; ++kk) {
      const int kbase = 4 * kk + koff;
      // A fragments for the two 16-row M-tiles (ds_load_b64 each)
      v2f a0 = *(const v2f*)(Yw + lane16 * WPITCH + kbase);
      v2f a1 = *(const v2f*)(Yw + (16 + lane16) * WPITCH + kbase);
      #pragma unroll
      for (int nt = 0; nt < 8; ++nt) {
        // B fragment fetched once, consumed by both M-tiles
        v2f b = *(const v2f*)(Wc + (nt * 16 + lane16) * WPITCH + kbase);
        acc[0][nt] = __builtin_amdgcn_wmma_f32_16x16x4_f32(
            false, a0, false, b, (short)0, acc[0][nt], false, false);
        acc[1][nt] = __builtin_amdgcn_wmma_f32_16x16x4_f32(
            false, a1, false, b, (short)0, acc[1][nt], false, false);
      }
    }

    // epilogue: Y_new[m][l] = x0[m]*P[m][l] + x1[m]*P[m][64+l]
    float* Ywr = Ybuf[wave];
    #pragma unroll
    for (int mt = 0; mt < 2; ++mt) {
      #pragma unroll
      for (int j = 0; j < 8; ++j) {
        const int row16 = j + half * 8;                  // D-row for VGPR j
        const int wgrow = wave * ROWS_PER_WAVE + mt * 16 + row16;
        const float x0 = xbuf[sb][wgrow];
        const float x1 = xbuf[sb][ROWS_PER_WG + wgrow];
        #pragma unroll
        for (int t = 0; t < 4; ++t) {
          float v = x0 * acc[mt][t][j] + x1 * acc[mt][t + 4][j];
          Ywr[(mt * 16 + row16) * WPITCH + t * 16 + lane16] = v;
        }
      }
    }
  }

  // ---- last site: out[n,c,o] = sum_{r,i} Y[n,r]*x[n,i]*w_last[r,i,o,c]
  __syncthreads();
  for (int t = tid; t < 2 * ROWS_PER_WG; t += BLOCK_THREADS)
    xbuf[0][t] = xT[((long long)(SITES - 1) * 2 + (t >> 6)) * NBATCH + n0 + (t & 63)];
  __syncthreads();

  for (int idx = tid; idx < ROWS_PER_WG * OUT_D; idx += BLOCK_THREADS) {
    int mg = idx / OUT_D;
    int o  = idx % OUT_D;
    float x0 = xbuf[0][mg];
    float x1 = xbuf[0][ROWS_PER_WG + mg];
    const float* Yr = Ybuf[mg >> 5] + (mg & 31) * WPITCH;
    float accv = 0.0f;
    for (int r = 0; r < HID; ++r) {
      float w0 = w_last[((r * 2 + 0) * OUT_D + o) * CH + c];
      float w1 = w_last[((r * 2 + 1) * OUT_D + o) * CH + c];
      accv += Yr[r] * (x0 * w0 + x1 * w1);
    }
    out[(long long)(n0 + mg) * (CH * OUT_D) + c * OUT_D + o] = accv;
  }
}

// ---------------------------------------------------------------------------
extern "C" void kernel_launch(void* const* d_in, const int* in_sizes, int n_in,
                              void* d_out, int out_size, void* d_ws, size_t ws_size,
                              hipStream_t stream) {
  (void)in_sizes; (void)n_in; (void)out_size; (void)ws_size;
  const float* x       = (const float*)d_in[0];
  const float* w_first = (const float*)d_in[1];
  const float* w_mid   = (const float*)d_in[2];
  const float* w_last  = (const float*)d_in[3];
  float* out = (float*)d_out;
  float* ws  = (float*)d_ws;   // requires ~27 MB

  const long long total = WP_TOTAL + XT_TOTAL;
  const int threads = 256;
  const int blocks  = (int)((total + threads - 1) / threads);
  TT_repack_kernel<<<blocks, threads, 0, stream>>>(x, w_mid, ws);

  TT_mps_chain_kernel<<<(NBATCH / ROWS_PER_WG) * CH, BLOCK_THREADS, 0, stream>>>(
      ws, w_first, w_last, out);
}